// AttentionBlock_12094627906384
// MI455X (gfx1250) — compile-verified
//
#include <hip/hip_runtime.h>
#include <math.h>

typedef __attribute__((ext_vector_type(2))) float v2f;
typedef __attribute__((ext_vector_type(8))) float v8f;
typedef int v4i __attribute__((vector_size(16)));

#define WMMA_F32(A, B, C) \
  __builtin_amdgcn_wmma_f32_16x16x4_f32(false, (A), false, (B), (short)0, (C), false, false)

// ---- async global->LDS copy (16 bytes per lane), with sync fallback --------
__device__ static inline void async_cp16(float* dst_lds, const float* src_glb) {
#if __has_builtin(__builtin_amdgcn_global_load_async_to_lds_b128)
  __builtin_amdgcn_global_load_async_to_lds_b128(
      (__attribute__((address_space(1))) v4i*)(src_glb),
      (__attribute__((address_space(3))) v4i*)(dst_lds), 0, 0);
#else
  *(float4*)dst_lds = *(const float4*)src_glb;
#endif
}

__device__ static inline void async_wait0() {
#if __has_builtin(__builtin_amdgcn_s_wait_asynccnt)
  __builtin_amdgcn_s_wait_asynccnt(0);
#else
  asm volatile("s_wait_asynccnt 0" ::: "memory");
#endif
}

// ---------------------------------------------------------------------------
// Kernel 1: GroupNorm  (b=8, c=512, t=1024, 32 groups of 16 channels)
// ---------------------------------------------------------------------------
__global__ void gn_kernel(const float* __restrict__ x, const float* __restrict__ gamma,
                          const float* __restrict__ beta, float* __restrict__ h) {
  const int bg = blockIdx.x;           // 256 blocks = 8 b * 32 g
  const int b = bg >> 5, g = bg & 31;
  const int T = 1024, CPG = 16, N = CPG * T;
  const float* xp = x + ((size_t)b * 512 + (size_t)g * CPG) * T;
  float* hp = h + ((size_t)b * 512 + (size_t)g * CPG) * T;
  const int tid = threadIdx.x;         // 256 threads

  float s = 0.f, ss = 0.f;
  for (int i = tid; i < N; i += 256) { float v = xp[i]; s += v; ss += v * v; }

  __shared__ float r0[256], r1[256];
  r0[tid] = s; r1[tid] = ss;
  __syncthreads();
  for (int off = 128; off > 0; off >>= 1) {
    if (tid < off) { r0[tid] += r0[tid + off]; r1[tid] += r1[tid + off]; }
    __syncthreads();
  }
  const float mean = r0[0] * (1.0f / 16384.0f);
  const float var  = r1[0] * (1.0f / 16384.0f) - mean * mean;
  const float rstd = rsqrtf(var + 1e-5f);

  for (int i = tid; i < N; i += 256) {
    int c = g * CPG + (i >> 10);
    hp[i] = (xp[i] - mean) * rstd * gamma[c] + beta[c];
  }
}

// ---------------------------------------------------------------------------
// Kernel 2: QKV GEMM, 2x2 wave tile (32 o x 32 t), 1.5 loads per WMMA.
// Writes q,k as (bh, ch, t); v transposed as (bh, t, ch).
// grid = (48 o-pairs, 8 t-macros(128), 8 b), block = 128 (4 waves)
// ---------------------------------------------------------------------------
__global__ void qkv_kernel(const float* __restrict__ h, const float* __restrict__ w,
                           const float* __restrict__ bias, float* __restrict__ qb,
                           float* __restrict__ kb, float* __restrict__ vtb) {
  const int lane = threadIdx.x & 31;
  const int wv   = threadIdx.x >> 5;
  const int o0 = blockIdx.x * 32;
  const int t0 = blockIdx.y * 128 + wv * 32;
  const int b  = blockIdx.z;
  const int l16 = lane & 15;
  const int kh  = (lane >> 4) << 1;
  const int half8 = (lane >> 4) << 3;

  const float* hp  = h + (size_t)b * 512 * 1024;
  const float* wr0 = w + (size_t)(o0 + l16) * 512 + kh;
  const float* wr1 = wr0 + (size_t)16 * 512;
  const float* hc0 = hp + (size_t)kh * 1024 + t0 + l16;
  const float* hc1 = hc0 + 16;

  v8f acc[2][2] = {};
  for (int c = 0; c < 512; c += 4) {
    v2f a0, a1, b0, b1;
    a0.x = wr0[c]; a0.y = wr0[c + 1];
    a1.x = wr1[c]; a1.y = wr1[c + 1];
    b0.x = hc0[(size_t)c * 1024]; b0.y = hc0[(size_t)c * 1024 + 1024];
    b1.x = hc1[(size_t)c * 1024]; b1.y = hc1[(size_t)c * 1024 + 1024];
    acc[0][0] = WMMA_F32(a0, b0, acc[0][0]);
    acc[0][1] = WMMA_F32(a0, b1, acc[0][1]);
    acc[1][0] = WMMA_F32(a1, b0, acc[1][0]);
    acc[1][1] = WMMA_F32(a1, b1, acc[1][1]);
  }

  // Routing is uniform per 16-row tile: a 16-aligned range never crosses a
  // 64/192 boundary, so hd/sec are scalar (SALU + uniform branch).
  for (int io = 0; io < 2; ++io) {
    const int ob  = o0 + io * 16;
    const int hd  = ob / 192;
    const int rem = ob - hd * 192;
    const int sec = rem >> 6;          // 0=q, 1=k, 2=v  (uniform)
    const int chb = rem & 63;
    const int bh  = b * 8 + hd;
    for (int it = 0; it < 2; ++it) {
      const int t = t0 + it * 16 + l16;
      for (int r = 0; r < 8; ++r) {
        const int ch  = chb + r + half8;
        const float val = acc[io][it][r] + bias[ob + r + half8];
        if (sec == 0)      qb[((size_t)bh * 64 + ch) * 1024 + t] = val;
        else if (sec == 1) kb[((size_t)bh * 64 + ch) * 1024 + t] = val;
        else               vtb[((size_t)bh * 1024 + t) * 64 + ch] = val;
      }
    }
  }
}

// ---------------------------------------------------------------------------
// Kernel 3: attention, online (flash) softmax, one wave per (bh, 16 queries).
// K/V tiles staged in LDS (double-buffered, async copies) shared by 4 waves.
// grid = (16, 64 bh), block = 128 (4 waves)
// ---------------------------------------------------------------------------
__global__ void attn_kernel(const float* __restrict__ qb, const float* __restrict__ kb,
                            const float* __restrict__ vtb, float* __restrict__ ab) {
  const int lane = threadIdx.x & 31;
  const int wv   = threadIdx.x >> 5;
  const int tid  = threadIdx.x;        // 0..127
  const int bh   = blockIdx.y;
  const int t0   = (blockIdx.x * 4 + wv) * 16;
  const int l16  = lane & 15;
  const int kh   = (lane >> 4) << 1;

  const float* q  = qb  + (size_t)bh * 64 * 1024;
  const float* k  = kb  + (size_t)bh * 64 * 1024;
  const float* vt = vtb + (size_t)bh * 1024 * 64;

  __shared__ float kt[2][64 * 16];     // [ch][s_local]
  __shared__ float vts[2][16 * 64];    // [s_local][ch]
  __shared__ float plds[4][16 * 16];
  float* myp = plds[wv];

  // Q A-fragments for all 16 K-steps (ch = 0..63), reused across all s-tiles.
  v2f qa[16];
  for (int kk = 0; kk < 16; ++kk) {
    int ch = 4 * kk + kh;
    qa[kk].x = q[(size_t)ch * 1024 + t0 + l16];
    qa[kk].y = q[(size_t)(ch + 1) * 1024 + t0 + l16];
  }

  const float s2 = 0.125f;             // scale^2 = 1/sqrt(ch) = 1/8

  auto copy_tiles = [&](int st, int buf) {
    const int s0 = st * 16;
    const float* ksrc = k + (size_t)(tid >> 1) * 1024 + s0 + (tid & 1) * 8;
    float* kdst = &kt[buf][(tid >> 1) * 16 + (tid & 1) * 8];
    async_cp16(kdst, ksrc);
    async_cp16(kdst + 4, ksrc + 4);
    const float* vsrc = vt + (size_t)s0 * 64 + tid * 8;
    float* vdst = &vts[buf][tid * 8];
    async_cp16(vdst, vsrc);
    async_cp16(vdst + 4, vsrc + 4);
  };

  float m[8], rs[8];
  for (int r = 0; r < 8; ++r) { m[r] = -1e30f; rs[r] = 0.f; }
  v8f O[4] = {};

  copy_tiles(0, 0);
  for (int st = 0; st < 64; ++st) {
    const int buf = st & 1;
    async_wait0();
    __syncthreads();                   // tile `st` visible; prev reads done
    if (st + 1 < 64) copy_tiles(st + 1, buf ^ 1);

    const float* kc = kt[buf];
    const float* vc = vts[buf];

    // S = (q^T k) tile
    v8f S = {};
    for (int kk = 0; kk < 16; ++kk) {
      int ch = 4 * kk + kh;
      v2f bb;
      bb.x = kc[ch * 16 + l16];
      bb.y = kc[(ch + 1) * 16 + l16];
      S = WMMA_F32(qa[kk], bb, S);
    }

    // online softmax update (rows live in 16-lane halves: M = r + 8*half)
    float fr[8];
    for (int r = 0; r < 8; ++r) {
      float sv = S[r] * s2;
      float tm = sv;
      for (int off = 1; off < 16; off <<= 1) tm = fmaxf(tm, __shfl_xor(tm, off, 32));
      float mn = fmaxf(m[r], tm);
      float f  = __expf(m[r] - mn);
      float p  = __expf(sv - mn);
      m[r]  = mn;
      rs[r] = rs[r] * f + p;
      fr[r] = f;
      myp[(r + ((lane >> 4) << 3)) * 16 + l16] = p;   // accum layout -> LDS
    }
    asm volatile("s_wait_dscnt 0" ::: "memory");

    // P as A-fragments
    v2f pa[4];
    for (int kk = 0; kk < 4; ++kk) {
      int sb = 4 * kk + kh;
      pa[kk] = *(const v2f*)(myp + l16 * 16 + sb);
    }

    // O = O*f + P @ V^T
    for (int ct = 0; ct < 4; ++ct) {
      v8f o = O[ct];
      for (int r = 0; r < 8; ++r) o[r] *= fr[r];
      const float* vb = vc + ct * 16 + l16;
      for (int kk = 0; kk < 4; ++kk) {
        int sl = 4 * kk + kh;
        v2f bb;
        bb.x = vb[sl * 64];
        bb.y = vb[(sl + 1) * 64];
        o = WMMA_F32(pa[kk], bb, o);
      }
      O[ct] = o;
    }
  }

  // reduce row sums and normalize
  for (int r = 0; r < 8; ++r) {
    float v = rs[r];
    for (int off = 1; off < 16; off <<= 1) v += __shfl_xor(v, off, 32);
    rs[r] = 1.0f / v;
  }

  // write a[b][hd*64+ch][t]
  const int b = bh >> 3, hd = bh & 7;
  float* ap = ab + ((size_t)b * 512 + (size_t)hd * 64) * 1024;
  for (int ct = 0; ct < 4; ++ct) {
    int ch = ct * 16 + l16;
    for (int r = 0; r < 8; ++r) {
      int tl = r + ((lane >> 4) << 3);
      ap[(size_t)ch * 1024 + t0 + tl] = O[ct][r] * rs[r];
    }
  }
}

// ---------------------------------------------------------------------------
// Kernel 4: proj GEMM + bias + residual, 2x2 wave tile.
// grid = (16 o-pairs, 8 t-macros, 8 b), block = 128
// ---------------------------------------------------------------------------
__global__ void proj_kernel(const float* __restrict__ a, const float* __restrict__ w,
                            const float* __restrict__ bias, const float* __restrict__ x,
                            float* __restrict__ out) {
  const int lane = threadIdx.x & 31;
  const int wv   = threadIdx.x >> 5;
  const int o0 = blockIdx.x * 32;
  const int t0 = blockIdx.y * 128 + wv * 32;
  const int b  = blockIdx.z;
  const int l16 = lane & 15;
  const int kh  = (lane >> 4) << 1;
  const int half8 = (lane >> 4) << 3;

  const float* ap  = a + (size_t)b * 512 * 1024;
  const float* wr0 = w + (size_t)(o0 + l16) * 512 + kh;
  const float* wr1 = wr0 + (size_t)16 * 512;
  const float* ac0 = ap + (size_t)kh * 1024 + t0 + l16;
  const float* ac1 = ac0 + 16;

  v8f acc[2][2] = {};
  for (int c = 0; c < 512; c += 4) {
    v2f a0, a1, b0, b1;
    a0.x = wr0[c]; a0.y = wr0[c + 1];
    a1.x = wr1[c]; a1.y = wr1[c + 1];
    b0.x = ac0[(size_t)c * 1024]; b0.y = ac0[(size_t)c * 1024 + 1024];
    b1.x = ac1[(size_t)c * 1024]; b1.y = ac1[(size_t)c * 1024 + 1024];
    acc[0][0] = WMMA_F32(a0, b0, acc[0][0]);
    acc[0][1] = WMMA_F32(a0, b1, acc[0][1]);
    acc[1][0] = WMMA_F32(a1, b0, acc[1][0]);
    acc[1][1] = WMMA_F32(a1, b1, acc[1][1]);
  }

  for (int io = 0; io < 2; ++io)
    for (int it = 0; it < 2; ++it)
      for (int r = 0; r < 8; ++r) {
        int o = o0 + io * 16 + r + half8;
        int t = t0 + it * 16 + l16;
        size_t idx = ((size_t)b * 512 + o) * 1024 + t;
        out[idx] = x[idx] + bias[o] + acc[io][it][r];
      }
}

// ---------------------------------------------------------------------------
extern "C" void kernel_launch(void* const* d_in, const int* in_sizes, int n_in,
                              void* d_out, int out_size, void* d_ws, size_t ws_size,
                              hipStream_t stream) {
  const float* x      = (const float*)d_in[0];
  const float* gamma  = (const float*)d_in[1];
  const float* beta   = (const float*)d_in[2];
  const float* w_qkv  = (const float*)d_in[3];
  const float* b_qkv  = (const float*)d_in[4];
  const float* w_proj = (const float*)d_in[5];
  const float* b_proj = (const float*)d_in[6];
  float* out = (float*)d_out;

  const size_t M4 = (size_t)4 * 1024 * 1024;   // 4M floats = 16 MB
  float* h   = (float*)d_ws;     // [8,512,1024]
  float* qb  = h  + M4;          // [64,64,1024]
  float* kb  = qb + M4;          // [64,64,1024]
  float* vtb = kb + M4;          // [64,1024,64]
  float* ab  = h;                // reuse h region for attention output

  gn_kernel  <<<256, 256, 0, stream>>>(x, gamma, beta, h);
  qkv_kernel <<<dim3(48, 8, 8), 128, 0, stream>>>(h, w_qkv, b_qkv, qb, kb, vtb);
  attn_kernel<<<dim3(16, 64),   128, 0, stream>>>(qb, kb, vtb, ab);
  proj_kernel<<<dim3(16, 8, 8), 128, 0, stream>>>(ab, w_proj, b_proj, x, out);
}